// ScaleAwareUpsampler_6330781794822
// MI455X (gfx1250) — compile-verified
//
#include <hip/hip_runtime.h>
#include <hip/hip_bf16.h>
#include <math.h>

typedef __attribute__((ext_vector_type(2))) float v2f;
typedef __attribute__((ext_vector_type(8))) float v8f;

#define IN_H 192
#define IN_W 192
#define OUT_H 384
#define OUT_W 384
#define NB 4
#define NC 64
#define NPIX (OUT_H * OUT_W)

__device__ __forceinline__ float gelu_exact(float v) {
    // jax.nn.gelu(approximate=False): 0.5*x*(1+erf(x/sqrt(2)))
    return 0.5f * v * (1.0f + erff(v * 0.70710678118654752440f));
}

// -------------------------------------------------------------------------
// Kernel 1: per-pixel coordinate MLP via f32 WMMA (16x16x4), producing
// routing[NPIX][4] (post-sigmoid) and offset[NPIX][2] into workspace.
// Block = 128 threads = 4 waves; each wave owns 16 pixels (one M-tile).
// -------------------------------------------------------------------------
__global__ __launch_bounds__(128) void moe_mlp_kernel(
    const float* __restrict__ w1, const float* __restrict__ b1,
    const float* __restrict__ w2, const float* __restrict__ b2,
    const float* __restrict__ rw, const float* __restrict__ rb,
    const float* __restrict__ off_w, const float* __restrict__ off_b,
    float* __restrict__ routing, float* __restrict__ offset)
{
    // padded stride 68 -> conflict-free float2 LDS access patterns
    __shared__ float sW2[64 * 68];
    __shared__ float sEmb[4 * 16 * 68];

    const int tid = threadIdx.x;
    for (int i = tid; i < 4096; i += 128)
        sW2[(i >> 6) * 68 + (i & 63)] = w2[i];
    __syncthreads();

    const int lane = tid & 31;
    const int wave = tid >> 5;
    const int m    = lane & 15;   // row within wave's 16-pixel tile
    const int half = lane >> 4;   // K-half per WMMA f32 layout
    const int pixBase = blockIdx.x * 64 + wave * 16;
    const int pix = pixBase + m;
    const int oh  = pix / OUT_W;
    const int owi = pix - oh * OUT_W;

    float* embW = sEmb + wave * (16 * 68);

    // ---- scale-aware coordinate features (scale = 2 -> 1/scale = 0.5) ----
    float chv = ((float)oh + 0.5f) * 0.5f;
    float coor_h = chv - floorf(chv + 1e-6f) - 0.5f;
    float cwv = ((float)owi + 0.5f) * 0.5f;
    float coor_w = cwv - floorf(cwv + 1e-6f) - 0.5f;
    // feat = [1/scale_w, 1/scale_h, coor_h, coor_w]
    v2f a1;
    a1.x = half ? coor_h : 0.5f;   // K = 2*half
    a1.y = half ? coor_w : 0.5f;   // K = 2*half + 1

    // ---- layer 1: [16x4] @ [4x64]  (one 16x16x4 WMMA per N-tile) ----
    v8f acc1[4];
    #pragma unroll
    for (int t = 0; t < 4; ++t) {
        const int n = t * 16 + m;
        v2f b;
        b.x = w1[n * 4 + 2 * half];       // B[k][n] = W1[n][k]
        b.y = w1[n * 4 + 2 * half + 1];
        v8f c = {};
        acc1[t] = __builtin_amdgcn_wmma_f32_16x16x4_f32(
            false, a1, false, b, (short)0, c, false, false);
    }
    // bias + gelu, scatter D-layout -> row-major emb1 in LDS
    #pragma unroll
    for (int t = 0; t < 4; ++t) {
        const float bias = b1[t * 16 + m];
        #pragma unroll
        for (int v = 0; v < 8; ++v)
            embW[(v + 8 * half) * 68 + t * 16 + m] = gelu_exact(acc1[t][v] + bias);
    }
    __syncthreads();

    // ---- layer 2: [16x64] @ [64x64]  (16 chained K=4 WMMAs per N-tile) ----
    v8f acc2[4];
    {
        v8f z = {};
        #pragma unroll
        for (int t = 0; t < 4; ++t) acc2[t] = z;
    }
    #pragma unroll
    for (int kt = 0; kt < 16; ++kt) {
        const int k0 = kt * 4 + 2 * half;
        v2f a;
        a.x = embW[m * 68 + k0];
        a.y = embW[m * 68 + k0 + 1];
        #pragma unroll
        for (int t = 0; t < 4; ++t) {
            const int n = t * 16 + m;
            v2f b;
            b.x = sW2[n * 68 + k0];       // B[k][n] = W2[n][k]
            b.y = sW2[n * 68 + k0 + 1];
            acc2[t] = __builtin_amdgcn_wmma_f32_16x16x4_f32(
                false, a, false, b, (short)0, acc2[t], false, false);
        }
    }
    __syncthreads();   // emb1 fully consumed before overwrite
    #pragma unroll
    for (int t = 0; t < 4; ++t) {
        const float bias = b2[t * 16 + m];
        #pragma unroll
        for (int v = 0; v < 8; ++v)
            embW[(v + 8 * half) * 68 + t * 16 + m] = gelu_exact(acc2[t][v] + bias);
    }
    __syncthreads();

    // ---- routing (sigmoid) + offset heads: lanes 0..15, one row each ----
    if (lane < 16) {
        float r0 = rb[0], r1 = rb[1], r2 = rb[2], r3 = rb[3];
        float o0 = off_b[0], o1 = off_b[1];
        const float* er = embW + m * 68;
        #pragma unroll 8
        for (int j = 0; j < 64; ++j) {
            const float e = er[j];
            r0 += e * rw[0 * 64 + j];
            r1 += e * rw[1 * 64 + j];
            r2 += e * rw[2 * 64 + j];
            r3 += e * rw[3 * 64 + j];
            o0 += e * off_w[0 * 64 + j];
            o1 += e * off_w[1 * 64 + j];
        }
        const int p = pixBase + m;
        routing[p * 4 + 0] = 1.0f / (1.0f + expf(-r0));
        routing[p * 4 + 1] = 1.0f / (1.0f + expf(-r1));
        routing[p * 4 + 2] = 1.0f / (1.0f + expf(-r2));
        routing[p * 4 + 3] = 1.0f / (1.0f + expf(-r3));
        offset[p * 2 + 0] = o0;
        offset[p * 2 + 1] = o1;
    }
}

// -------------------------------------------------------------------------
// Kernel 2: fused bilinear grid-sample (learned offsets, zero padding) +
// per-pixel rank-4 bottleneck + residual. One thread per (pixel, batch),
// 64-channel loop fully unrolled (fea in VGPRs). Expert tables transposed
// into LDS as [c][e][r] so per-channel weights are broadcast b128 DS loads.
// -------------------------------------------------------------------------
__global__ __launch_bounds__(256) void upsample_apply_kernel(
    const float* __restrict__ x, const float* __restrict__ wcomp,
    const float* __restrict__ wexp, const float* __restrict__ routing,
    const float* __restrict__ offset, float* __restrict__ out)
{
    __shared__ __align__(16) float sWC[1024];  // [c][e][r] of WC[e][r][c]
    __shared__ __align__(16) float sWE[1024];  // [c][e][r] of WE[e][c][r]
    const int tid = threadIdx.x;
    for (int i = tid; i < 1024; i += 256) {
        const int e = i >> 8;
        const int rA = (i >> 6) & 3, cA = i & 63;          // wcomp[e][rA][cA]
        sWC[cA * 16 + e * 4 + rA] = wcomp[i];
        const int cB = (i >> 2) & 63, rB = i & 3;          // wexp[e][cB][rB]
        sWE[cB * 16 + e * 4 + rB] = wexp[i];
    }
    __syncthreads();

    const int pix = blockIdx.x * 256 + tid;
    const int b   = blockIdx.y;
    const int oh  = pix / OUT_W;
    const int owi = pix - oh * OUT_W;

    const float offx = offset[pix * 2 + 0];
    const float offy = offset[pix * 2 + 1];

    const float inv_wm1 = 1.0f / (float)(IN_W - 1);
    const float inv_hm1 = 1.0f / (float)(IN_H - 1);
    float gx = (((float)owi + 0.5f) * 0.5f - 0.5f) * 2.0f * inv_wm1 - 1.0f;
    float gy = (((float)oh  + 0.5f) * 0.5f - 0.5f) * 2.0f * inv_hm1 - 1.0f;
    gx += offx * 2.0f * inv_wm1;
    gy += offy * 2.0f * inv_hm1;
    const float ixf = (gx + 1.0f) * ((float)IN_W * 0.5f) - 0.5f;
    const float iyf = (gy + 1.0f) * ((float)IN_H * 0.5f) - 0.5f;
    const float x0f = floorf(ixf), y0f = floorf(iyf);
    const float x1f = x0f + 1.0f,  y1f = y0f + 1.0f;
    const float wx1 = ixf - x0f, wx0 = 1.0f - wx1;
    const float wy1 = iyf - y0f, wy0 = 1.0f - wy1;
    const float vx0 = (x0f >= 0.0f && x0f <= (float)(IN_W - 1)) ? 1.0f : 0.0f;
    const float vx1 = (x1f >= 0.0f && x1f <= (float)(IN_W - 1)) ? 1.0f : 0.0f;
    const float vy0 = (y0f >= 0.0f && y0f <= (float)(IN_H - 1)) ? 1.0f : 0.0f;
    const float vy1 = (y1f >= 0.0f && y1f <= (float)(IN_H - 1)) ? 1.0f : 0.0f;
    const int xi0 = min(max((int)x0f, 0), IN_W - 1);
    const int xi1 = min(max((int)x1f, 0), IN_W - 1);
    const int yi0 = min(max((int)y0f, 0), IN_H - 1);
    const int yi1 = min(max((int)y1f, 0), IN_H - 1);
    const float w00 = wy0 * wx0 * vy0 * vx0;
    const float w01 = wy0 * wx1 * vy0 * vx1;
    const float w10 = wy1 * wx0 * vy1 * vx0;
    const float w11 = wy1 * wx1 * vy1 * vx1;
    const int i00 = yi0 * IN_W + xi0, i01 = yi0 * IN_W + xi1;
    const int i10 = yi1 * IN_W + xi0, i11 = yi1 * IN_W + xi1;

    const float r0 = routing[pix * 4 + 0];
    const float r1 = routing[pix * 4 + 1];
    const float r2 = routing[pix * 4 + 2];
    const float r3 = routing[pix * 4 + 3];

    const float* xb = x + (size_t)b * NC * (IN_H * IN_W);
    float fea[NC];
    float mid0 = 0.f, mid1 = 0.f, mid2 = 0.f, mid3 = 0.f;
    #pragma unroll
    for (int c = 0; c < NC; ++c) {
        const float* xc = xb + c * (IN_H * IN_W);
        const float f = w00 * xc[i00] + w01 * xc[i01] + w10 * xc[i10] + w11 * xc[i11];
        fea[c] = f;
        const float4 q0 = *(const float4*)&sWC[c * 16 + 0];
        const float4 q1 = *(const float4*)&sWC[c * 16 + 4];
        const float4 q2 = *(const float4*)&sWC[c * 16 + 8];
        const float4 q3 = *(const float4*)&sWC[c * 16 + 12];
        mid0 += (r0 * q0.x + r1 * q1.x + r2 * q2.x + r3 * q3.x) * f;
        mid1 += (r0 * q0.y + r1 * q1.y + r2 * q2.y + r3 * q3.y) * f;
        mid2 += (r0 * q0.z + r1 * q1.z + r2 * q2.z + r3 * q3.z) * f;
        mid3 += (r0 * q0.w + r1 * q1.w + r2 * q2.w + r3 * q3.w) * f;
    }

    float* outp = out + (size_t)b * NC * NPIX + (size_t)oh * OUT_W + owi;
    #pragma unroll
    for (int c = 0; c < NC; ++c) {
        const float4 q0 = *(const float4*)&sWE[c * 16 + 0];
        const float4 q1 = *(const float4*)&sWE[c * 16 + 4];
        const float4 q2 = *(const float4*)&sWE[c * 16 + 8];
        const float4 q3 = *(const float4*)&sWE[c * 16 + 12];
        float o = fea[c];
        o += (r0 * q0.x + r1 * q1.x + r2 * q2.x + r3 * q3.x) * mid0;
        o += (r0 * q0.y + r1 * q1.y + r2 * q2.y + r3 * q3.y) * mid1;
        o += (r0 * q0.z + r1 * q1.z + r2 * q2.z + r3 * q3.z) * mid2;
        o += (r0 * q0.w + r1 * q1.w + r2 * q2.w + r3 * q3.w) * mid3;
        outp[(size_t)c * NPIX] = o;
    }
}

extern "C" void kernel_launch(void* const* d_in, const int* in_sizes, int n_in,
                              void* d_out, int out_size, void* d_ws, size_t ws_size,
                              hipStream_t stream) {
    (void)in_sizes; (void)n_in; (void)out_size; (void)ws_size;
    const float* x     = (const float*)d_in[0];
    const float* wcomp = (const float*)d_in[1];
    const float* wexp  = (const float*)d_in[2];
    const float* w1    = (const float*)d_in[3];
    const float* b1    = (const float*)d_in[4];
    const float* w2    = (const float*)d_in[5];
    const float* b2    = (const float*)d_in[6];
    const float* rw    = (const float*)d_in[7];
    const float* rb    = (const float*)d_in[8];
    const float* offw  = (const float*)d_in[9];
    const float* offb  = (const float*)d_in[10];
    float* out = (float*)d_out;

    float* routing = (float*)d_ws;            // NPIX*4 floats
    float* offset  = routing + (size_t)NPIX * 4;  // NPIX*2 floats (ws total ~3.4 MB)

    moe_mlp_kernel<<<NPIX / 64, 128, 0, stream>>>(
        w1, b1, w2, b2, rw, rb, offw, offb, routing, offset);
    upsample_apply_kernel<<<dim3(NPIX / 256, NB), 256, 0, stream>>>(
        x, wcomp, wexp, routing, offset, out);
}